// coarse_heatmap_54752243089504
// MI455X (gfx1250) — compile-verified
//
#include <hip/hip_runtime.h>
#include <math.h>

// Problem constants (from reference)
#define Dd 96
#define Hh 96
#define Ww 96
#define NL 19
#define BB 4
#define BN (BB * NL)          // 76 (b,n) pairs
#define SLICE (Hh * Ww)       // 9216 elements per (b,n,d) slice
#define NBLK (BN * Dd)        // 7296 slices / blocks
#define INV_2SIG2 (-0.125f)   // -1/(2*sigma^2), sigma=2

typedef __attribute__((ext_vector_type(2))) float v2f;
typedef __attribute__((ext_vector_type(8))) float v8f;

// ---------------------------------------------------------------------------
// Kernel 1: per-(b,n) separable Gaussian tables ex/ey/ez (96 each), the
// normalization 1/(Σex·Σey·Σez), and the validity mask. 76 blocks x 128 thr.
// ---------------------------------------------------------------------------
__global__ __launch_bounds__(128) void tables_kernel(
    const float* __restrict__ labels,
    float* __restrict__ ex, float* __restrict__ ey, float* __restrict__ ez,
    float* __restrict__ invS, float* __restrict__ maskArr)
{
    const int bn = blockIdx.x;
    const int i  = threadIdx.x;          // 0..127 (only 0..95 compute)
    const float lx = labels[bn * 3 + 0];
    const float ly = labels[bn * 3 + 1];
    const float lz = labels[bn * 3 + 2];
    const float x = lx * (float)(Ww - 1);
    const float y = ly * (float)(Hh - 1);
    const float z = lz * (float)(Dd - 1);

    float vx = 0.0f, vy = 0.0f, vz = 0.0f;
    if (i < 96) {
        const float fi = (float)i;
        const float dx = fi - x, dy = fi - y, dz = fi - z;
        vx = __expf(dx * dx * INV_2SIG2);   // v_exp_f32
        vy = __expf(dy * dy * INV_2SIG2);
        vz = __expf(dz * dz * INV_2SIG2);
        ex[bn * 96 + i] = vx;
        ey[bn * 96 + i] = vy;
        ez[bn * 96 + i] = vz;
    }

    __shared__ float sx[128], sy[128], sz[128];
    sx[i] = vx; sy[i] = vy; sz[i] = vz;
    __syncthreads();
    for (int s = 64; s > 0; s >>= 1) {
        if (i < s) { sx[i] += sx[i + s]; sy[i] += sy[i + s]; sz[i] += sz[i + s]; }
        __syncthreads();
    }
    if (i == 0) {
        const float S = sx[0] * sy[0] * sz[0];
        invS[bn]    = (S > 0.0f) ? (1.0f / S) : 1.0f;   // reference s>0 guard
        maskArr[bn] = (lx >= 0.0f) ? 1.0f : 0.0f;
    }
}

// ---------------------------------------------------------------------------
// Kernel 2: HBM-bound streaming L1 accumulation. One block per (b,n,d) slice
// (9216 voxels). Each thread: 9 x b128 loads of pred, target rebuilt from LDS
// 1-D tables. Wave partial sums folded with V_WMMA_F32_16X16X4_F32 (B = ones)
// then a block LDS reduce produces one partial per block.
// ---------------------------------------------------------------------------
__global__ __launch_bounds__(256) void loss_kernel(
    const float* __restrict__ pred,
    const float* __restrict__ exT, const float* __restrict__ eyT,
    const float* __restrict__ ezT, const float* __restrict__ invS,
    const float* __restrict__ maskArr,
    float* __restrict__ partials)
{
    __shared__ __align__(16) float s_ex[96];
    __shared__ float s_ey[96];
    __shared__ float red[256];

    const int bid = blockIdx.x;          // == (b*NL + n)*Dd + d
    const int bn  = bid / Dd;
    const int d   = bid - bn * Dd;
    const int tid = threadIdx.x;

    if (tid < 96)        s_ex[tid]      = exT[bn * 96 + tid];
    else if (tid < 192)  s_ey[tid - 96] = eyT[bn * 96 + (tid - 96)];
    __syncthreads();

    const float c0 = ezT[bn * 96 + d] * invS[bn];   // ez[d] / S, uniform
    const float m  = maskArr[bn];                   // 0/1, uniform
    const float* base = pred + (size_t)bid * SLICE;

    float accx = 0.0f, accy = 0.0f;
#pragma unroll
    for (int it = 0; it < 9; ++it) {
        const int f = tid * 4 + it * 1024;          // 16B-aligned, < 9216
        if (it < 8)                                  // global_prefetch_b8
            __builtin_prefetch(base + f + 1024, 0, 0);
        const float4 p = *reinterpret_cast<const float4*>(base + f);
        const int h = f / 96;                        // mul-shift
        const int w = f - h * 96;                    // multiple of 4
        const float  cy = c0 * s_ey[h];
        const float4 e  = *reinterpret_cast<const float4*>(&s_ex[w]);
        accx += fabsf(p.x - cy * e.x) + fabsf(p.y - cy * e.y);
        accy += fabsf(p.z - cy * e.z) + fabsf(p.w - cy * e.w);
    }
    accx *= m;
    accy *= m;

    // --- WMMA f32 reduction: D[m][n] = sum_k A[m][k] * 1 + 0 -------------
    // A (16x4 f32): lane L<16 holds A[L][0..1], lane 16+L holds A[L][2..3].
    // Sum of all 32 lanes' accumulators == sum_m rowsum[m].
    v2f a; a[0] = accx; a[1] = accy;
    v2f b; b[0] = 1.0f; b[1] = 1.0f;                // ones matrix (any layout)
    v8f c = {};
    v8f dm = __builtin_amdgcn_wmma_f32_16x16x4_f32(
        /*neg_a=*/false, a, /*neg_b=*/false, b,
        /*c_mod=*/(short)0, c, /*reuse_a=*/false, /*reuse_b=*/false);
    // Per lane: 8 rows of one column -> rows 0..7 (lanes 0-15) or 8..15
    // (lanes 16-31); identical within each half, so scale by 1/16 and let the
    // block reduce count every lane once.
    float s8 = dm[0] + dm[1] + dm[2] + dm[3] + dm[4] + dm[5] + dm[6] + dm[7];
    red[tid] = s8 * 0.0625f;
    __syncthreads();
    for (int s = 128; s > 0; s >>= 1) {
        if (tid < s) red[tid] += red[tid + s];
        __syncthreads();
    }
    if (tid == 0) partials[bid] = red[0];
}

// ---------------------------------------------------------------------------
// Kernel 3: reduce the 7296 block partials + 76 masks, write the scalar loss.
// ---------------------------------------------------------------------------
__global__ __launch_bounds__(256) void finalize_kernel(
    const float* __restrict__ partials, const float* __restrict__ maskArr,
    float* __restrict__ out)
{
    __shared__ float red[256];
    const int tid = threadIdx.x;
    float s = 0.0f;
    for (int i = tid; i < NBLK; i += 256) s += partials[i];
    red[tid] = s;
    __syncthreads();
    for (int st = 128; st > 0; st >>= 1) {
        if (tid < st) red[tid] += red[tid + st];
        __syncthreads();
    }
    if (tid == 0) {
        float vc = 0.0f;
        for (int i = 0; i < BN; ++i) vc += maskArr[i];
        out[0] = red[0] / (vc + 1e-8f);
    }
}

// ---------------------------------------------------------------------------
extern "C" void kernel_launch(void* const* d_in, const int* in_sizes, int n_in,
                              void* d_out, int out_size, void* d_ws, size_t ws_size,
                              hipStream_t stream) {
    (void)in_sizes; (void)n_in; (void)out_size; (void)ws_size;
    const float* pred   = (const float*)d_in[0];   // (4,19,96,96,96) f32
    const float* labels = (const float*)d_in[1];   // (4,19,3) f32
    float* out = (float*)d_out;                    // scalar f32

    // Workspace layout (floats): ~117 KB total
    float* ws       = (float*)d_ws;
    float* exT      = ws;                  // BN*96
    float* eyT      = exT + BN * 96;       // BN*96
    float* ezT      = eyT + BN * 96;       // BN*96
    float* invS     = ezT + BN * 96;       // BN
    float* maskArr  = invS + BN;           // BN
    float* partials = maskArr + BN;        // NBLK

    tables_kernel<<<BN, 128, 0, stream>>>(labels, exT, eyT, ezT, invS, maskArr);
    loss_kernel<<<NBLK, 256, 0, stream>>>(pred, exT, eyT, ezT, invS, maskArr, partials);
    finalize_kernel<<<1, 256, 0, stream>>>(partials, maskArr, out);
}